// MPNN_LSTM_18889266168016
// MI455X (gfx1250) — compile-verified
//
#include <hip/hip_runtime.h>

typedef __attribute__((ext_vector_type(2))) float        v2f;
typedef __attribute__((ext_vector_type(8))) float        v8f;
typedef __attribute__((ext_vector_type(4))) unsigned int v4u;
typedef __attribute__((ext_vector_type(4))) int          v4i;
typedef __attribute__((ext_vector_type(8))) int          v8i;

#define NBATCH 4
#define WINLEN 10
#define NNODE  10000
#define NFEAT  8
#define NHID   64
#define NTOT   (NBATCH*WINLEN*NNODE)   /* 400000 stacked nodes  */
#define NSEQ   (NBATCH*NNODE)          /* 40000 sequence batch  */
#define NEDGE  3200000
#define KMAX   208                     /* largest GEMM K (fc1)  */

#if defined(__has_builtin)
#  if __has_builtin(__builtin_amdgcn_tensor_load_to_lds)
#    define USE_TDM 1
#  endif
#endif
#if __has_include(<hip/amd_detail/amd_gfx1250_TDM.h>)
#  define TDM_ARGS6 1
#endif

static __device__ __forceinline__ float sigmoidf_(float x) {
  return 1.0f / (1.0f + expf(-x));
}

// ---------------------------------------------------------------- utilities
__global__ void k_fill(float* __restrict__ p, float v, long n) {
  long i = (long)blockIdx.x * blockDim.x + threadIdx.x;
  if (i < n) p[i] = v;
}

// ------------------------------------------------------- WMMA f32 GEMM core
// C[M,N] = A[M,K] @ W[K,N] (+bias, optional relu). Row-major everywhere.
// M % 32 == 0, K % 4 == 0 (K <= KMAX), N % 64 == 0.
// grid = (ceil(M/32/8), N/64); block = 256 (8 waves); wave -> 32x64 strip.
// The block's K x 64 weight strip is staged into LDS once (TDM if available);
// B fragments are read back with explicit ds_load_b32 (opaque to alias
// analysis, since the TDM write is invisible to the compiler).
__global__ __launch_bounds__(256) void k_gemm(
    const float* __restrict__ A, const float* __restrict__ W,
    const float* __restrict__ bias, float* __restrict__ C,
    int M, int K, int N, int relu)
{
  __shared__ float sw[KMAX * 64];
  int wv   = threadIdx.x >> 5;
  int lane = threadIdx.x & 31;
  int n0   = blockIdx.y << 6;
  // hardware LDS byte address of sw (flat address low 32 bits, ISA 10.2)
  unsigned swoff = (unsigned)(unsigned long long)(size_t)&sw[0];

  // ---- stage W[0:K][n0:n0+64] -> sw[K][64] -------------------------------
#ifdef USE_TDM
  if (wv == 0) {
    unsigned long long ga = (unsigned long long)(const char*)(W + n0);
    v4u g0;
    g0.x = 1u;                                   // count=1, user descriptor
    g0.y = swoff;                                // lds_addr (bytes)
    g0.z = (unsigned)ga;                         // global_addr[31:0]
    g0.w = (unsigned)((ga >> 32) & 0x01ffffffULL) | (2u << 30); // [56:32]|type=2
    const int td = 1 << 20;                      // oversized dims: no OOB clip
    v8i g1;
    g1.s0 = 0x00020000;                          // wg_mask=0, data_size=4B
    g1.s1 = (td & 0xffff) << 16;                 // tensor_dim0[15:0]
    g1.s2 = (td >> 16) | ((td & 0xffff) << 16);  // dim0[31:16] | dim1[15:0]
    g1.s3 = (td >> 16) | (64 << 16);             // dim1[31:16] | tile_dim0=64
    g1.s4 = (K & 0xffff);                        // tile_dim1=K, tile_dim2=0
    g1.s5 = N;                                   // tensor_dim0_stride[31:0]
    g1.s6 = 0;                                   // stride0 hi | stride1 lo
    g1.s7 = 0;
    v4i gz = {0, 0, 0, 0};
#  ifdef TDM_ARGS6
    v8i gz8 = {0, 0, 0, 0, 0, 0, 0, 0};
    __builtin_amdgcn_tensor_load_to_lds(g0, g1, gz, gz, gz8, 0);
#  else
    __builtin_amdgcn_tensor_load_to_lds(g0, g1, gz, gz, 0);
#  endif
    __builtin_amdgcn_s_wait_tensorcnt(0);
  }
#else
  for (int idx = threadIdx.x; idx < (K << 6); idx += 256) {
    int kr = idx >> 6, cc = idx & 63;
    sw[idx] = W[(long)kr * N + n0 + cc];
  }
#endif
  __syncthreads();

  int mt = blockIdx.x * 8 + wv;              // 32-row strip id
  int mtiles = M >> 5;
  if (mt >= mtiles) return;                  // wave-uniform: EXEC stays all-1s
  int m0 = mt << 5;
  int lr = lane & 15;                        // A row / B,C col within tile
  int kh = (lane >> 4) << 1;                 // K sub-offset: 0 or 2

  v8f c00 = {}; v8f c01 = {}; v8f c02 = {}; v8f c03 = {};
  v8f c10 = {}; v8f c11 = {}; v8f c12 = {}; v8f c13 = {};
  const float* Ap0 = A + (long)(m0 + lr) * K + kh;
  const float* Ap1 = Ap0 + (long)16 * K;
  unsigned sbase = swoff + (((unsigned)(kh << 6) + (unsigned)lr) << 2);

  for (int k = 0; k < K; k += 4) {
    v2f a0 = *(const v2f*)(Ap0 + k);         // rows m0..m0+15
    v2f a1 = *(const v2f*)(Ap1 + k);         // rows m0+16..m0+31
    unsigned ad = sbase + ((unsigned)k << 8);
    float b0x, b0y, b1x, b1y, b2x, b2y, b3x, b3y;
    asm volatile(
      "ds_load_b32 %0, %8 offset:0\n\t"
      "ds_load_b32 %1, %8 offset:256\n\t"
      "ds_load_b32 %2, %8 offset:64\n\t"
      "ds_load_b32 %3, %8 offset:320\n\t"
      "ds_load_b32 %4, %8 offset:128\n\t"
      "ds_load_b32 %5, %8 offset:384\n\t"
      "ds_load_b32 %6, %8 offset:192\n\t"
      "ds_load_b32 %7, %8 offset:448\n\t"
      "s_wait_dscnt 0x0"
      : "=&v"(b0x), "=&v"(b0y), "=&v"(b1x), "=&v"(b1y),
        "=&v"(b2x), "=&v"(b2y), "=&v"(b3x), "=&v"(b3y)
      : "v"(ad));
    v2f b0 = {b0x, b0y}, b1 = {b1x, b1y}, b2 = {b2x, b2y}, b3 = {b3x, b3y};
    c00 = __builtin_amdgcn_wmma_f32_16x16x4_f32(false, a0, false, b0, (short)0, c00, false, false);
    c01 = __builtin_amdgcn_wmma_f32_16x16x4_f32(false, a0, false, b1, (short)0, c01, false, false);
    c02 = __builtin_amdgcn_wmma_f32_16x16x4_f32(false, a0, false, b2, (short)0, c02, false, false);
    c03 = __builtin_amdgcn_wmma_f32_16x16x4_f32(false, a0, false, b3, (short)0, c03, false, false);
    c10 = __builtin_amdgcn_wmma_f32_16x16x4_f32(false, a1, false, b0, (short)0, c10, false, false);
    c11 = __builtin_amdgcn_wmma_f32_16x16x4_f32(false, a1, false, b1, (short)0, c11, false, false);
    c12 = __builtin_amdgcn_wmma_f32_16x16x4_f32(false, a1, false, b2, (short)0, c12, false, false);
    c13 = __builtin_amdgcn_wmma_f32_16x16x4_f32(false, a1, false, b3, (short)0, c13, false, false);
  }

  float bb0 = 0.f, bb1 = 0.f, bb2 = 0.f, bb3 = 0.f;
  if (bias) { bb0 = bias[n0+lr]; bb1 = bias[n0+lr+16]; bb2 = bias[n0+lr+32]; bb3 = bias[n0+lr+48]; }
  int r0 = m0 + ((lane >> 4) << 3);          // D layout: VGPR v -> row v / v+8
  #pragma unroll
  for (int v = 0; v < 8; ++v) {
    float o0 = c00[v] + bb0, o1 = c01[v] + bb1, o2 = c02[v] + bb2, o3 = c03[v] + bb3;
    float p0 = c10[v] + bb0, p1 = c11[v] + bb1, p2 = c12[v] + bb2, p3 = c13[v] + bb3;
    if (relu) {
      o0 = fmaxf(o0,0.f); o1 = fmaxf(o1,0.f); o2 = fmaxf(o2,0.f); o3 = fmaxf(o3,0.f);
      p0 = fmaxf(p0,0.f); p1 = fmaxf(p1,0.f); p2 = fmaxf(p2,0.f); p3 = fmaxf(p3,0.f);
    }
    float* Cp = C + (long)(r0 + v) * N + n0 + lr;
    Cp[0] = o0; Cp[16] = o1; Cp[32] = o2; Cp[48] = o3;
    float* Cq = Cp + (long)16 * N;
    Cq[0] = p0; Cq[16] = p1; Cq[32] = p2; Cq[48] = p3;
  }
}

// ------------------------------------------------------------ GCN machinery
__global__ void k_deg(const int* __restrict__ col, const float* __restrict__ ew,
                      float* __restrict__ deg) {
  long e = (long)blockIdx.x * blockDim.x + threadIdx.x;
  if (e < NEDGE) atomicAdd(&deg[col[e]], ew[e]);
}

__global__ void k_dinv(float* __restrict__ deg) {
  long i = (long)blockIdx.x * blockDim.x + threadIdx.x;
  if (i < NTOT) { float d = deg[i]; deg[i] = d > 0.f ? rsqrtf(d) : 0.f; }
}

// self-loop term: agg[v][h] = dinv[v]^2 * xl[v][h]  (also initializes agg)
__global__ void k_selfloop(const float* __restrict__ xl, const float* __restrict__ dinv,
                           float* __restrict__ agg) {
  long t = (long)blockIdx.x * blockDim.x + threadIdx.x;
  if (t >= (long)NTOT * NHID) return;
  float di = dinv[t >> 6];
  agg[t] = di * di * xl[t];
}

// one thread per (edge, 4 features): agg[col] += dinv[row]*w*dinv[col]*xl[row]
__global__ void k_edge_scatter(const int* __restrict__ row, const int* __restrict__ col,
                               const float* __restrict__ ew, const float* __restrict__ dinv,
                               const float* __restrict__ xl, float* __restrict__ agg) {
  long t = (long)blockIdx.x * blockDim.x + threadIdx.x;
  int e = (int)(t >> 4);
  int q = ((int)t & 15) << 2;
  if (e >= NEDGE) return;
  int r = row[e], c = col[e];
  float nrm = dinv[r] * ew[e] * dinv[c];
  float4 xv = *(const float4*)(xl + (long)r * NHID + q);
  float* ap = agg + (long)c * NHID + q;
  atomicAdd(ap + 0, nrm * xv.x);
  atomicAdd(ap + 1, nrm * xv.y);
  atomicAdd(ap + 2, nrm * xv.z);
  atomicAdd(ap + 3, nrm * xv.w);
}

// y = relu(agg + b); accumulate f64 sum & sumsq per feature (block pre-reduce)
__global__ __launch_bounds__(256) void k_bias_relu_stats(
    const float* __restrict__ agg, const float* __restrict__ bias,
    float* __restrict__ y, double* __restrict__ stats) {
  __shared__ double sh[2][256];
  int t = threadIdx.x;
  int h = t & 63;
  int rg = t >> 6;                               // 0..3
  float bb = bias[h];
  double s = 0.0, s2 = 0.0;
  for (long r = (long)blockIdx.x * 4 + rg; r < NTOT; r += (long)gridDim.x * 4) {
    float v = fmaxf(agg[r * NHID + h] + bb, 0.f);
    y[r * NHID + h] = v;
    s += v; s2 += (double)v * (double)v;
  }
  sh[0][t] = s; sh[1][t] = s2;
  __syncthreads();
  if (rg == 0) {
    double ts  = sh[0][h] + sh[0][64+h] + sh[0][128+h] + sh[0][192+h];
    double ts2 = sh[1][h] + sh[1][64+h] + sh[1][128+h] + sh[1][192+h];
    atomicAdd(&stats[h], ts);
    atomicAdd(&stats[64 + h], ts2);
  }
}

// training-mode BN: y = (y - m) * g*rsqrt(var+eps) + beta
__global__ void k_bn_apply(float* __restrict__ y, const double* __restrict__ stats,
                           const float* __restrict__ g, const float* __restrict__ bta) {
  long t = (long)blockIdx.x * blockDim.x + threadIdx.x;
  if (t >= (long)NTOT * NHID) return;
  int h = (int)(t & 63);
  double m   = stats[h] * (1.0 / (double)NTOT);
  double var = stats[64 + h] * (1.0 / (double)NTOT) - m * m;
  float sc = g[h] * rsqrtf((float)var + 1e-5f);
  y[t] = ((y[t] - (float)m) * sc) + bta[h];
}

// ------------------------------------------------------------ LSTM machinery
// pack [wih | whh] -> K-major Wcat[Kin+64][256]; bcat = bih + bhh
__global__ void k_pack_lstm_w(const float* __restrict__ wih, const float* __restrict__ whh,
                              const float* __restrict__ bih, const float* __restrict__ bhh,
                              float* __restrict__ Wcat, float* __restrict__ bcat, int Kin) {
  int tid = blockIdx.x * blockDim.x + threadIdx.x;
  if (tid < 256) bcat[tid] = bih[tid] + bhh[tid];
  int total = (Kin + 64) * 256;
  if (tid >= total) return;
  int k = tid / 256, j = tid % 256;
  Wcat[tid] = (k < Kin) ? wih[j * Kin + k] : whh[j * 64 + (k - Kin)];
}

__global__ void k_pack_fc1w(const float* __restrict__ fc1w, float* __restrict__ WT) {
  int tid = blockIdx.x * blockDim.x + threadIdx.x;   // k*64 + n
  if (tid >= 208 * 64) return;
  int k = tid >> 6, n = tid & 63;
  WT[tid] = fc1w[n * 208 + k];
}

// A_t = [ xc_t | h1state ]  (xc_t cols: h1 then h2 at remapped time-major row)
__global__ void k_gather_lstm1(const float* __restrict__ h1, const float* __restrict__ h2,
                               const float* __restrict__ hstate, float* __restrict__ Astep, int t) {
  int n = blockIdx.x;              // 0..NSEQ-1
  int k = threadIdx.x;             // 0..191
  int b = n / NNODE, i = n % NNODE;
  long r = ((long)(b * WINLEN + t) * NNODE + i);
  float v;
  if (k < 64)       v = h1[r * NHID + k];
  else if (k < 128) v = h2[r * NHID + (k - 64)];
  else              v = hstate[(long)n * NHID + (k - 128)];
  Astep[(long)n * 192 + k] = v;
}

__global__ void k_gather_lstm2(const float* __restrict__ y1, const float* __restrict__ hstate,
                               float* __restrict__ Astep) {
  int n = blockIdx.x;
  int k = threadIdx.x;             // 0..127
  float v = (k < 64) ? y1[(long)n * NHID + k] : hstate[(long)n * NHID + (k - 64)];
  Astep[(long)n * 128 + k] = v;
}

// gate order i,f,g,o ; c = sig(f)*c + sig(i)*tanh(g); h = sig(o)*tanh(c)
__global__ void k_lstm_cell(const float* __restrict__ gates,
                            float* __restrict__ c, float* __restrict__ h) {
  long t = (long)blockIdx.x * blockDim.x + threadIdx.x;
  if (t >= (long)NSEQ * NHID) return;
  long n = t >> 6; int j = (int)(t & 63);
  const float* g = gates + n * 256;
  float ig = sigmoidf_(g[j]);
  float fg = sigmoidf_(g[64 + j]);
  float gg = tanhf(g[128 + j]);
  float og = sigmoidf_(g[192 + j]);
  float cv = fg * c[t] + ig * gg;
  c[t] = cv;
  h[t] = og * tanhf(cv);
}

// ------------------------------------------------------------------- head
// Z = [hn1 | hn2 | skip] ; skip[n][w*8+f] = x[((b*WIN+w)*NN+i)*8+f]
__global__ void k_gather_z(const float* __restrict__ hn1, const float* __restrict__ hn2,
                           const float* __restrict__ x, float* __restrict__ Z) {
  int n = blockIdx.x;
  int k = threadIdx.x;
  if (k >= 208) return;
  float v;
  if (k < 64)       v = hn1[(long)n * NHID + k];
  else if (k < 128) v = hn2[(long)n * NHID + (k - 64)];
  else {
    int s = k - 128, w = s >> 3, f = s & 7;
    int b = n / NNODE, i = n % NNODE;
    v = x[((long)(b * WINLEN + w) * NNODE + i) * NFEAT + f];
  }
  Z[(long)n * 208 + k] = v;
}

__global__ void k_fc2(const float* __restrict__ a, const float* __restrict__ w,
                      const float* __restrict__ b, float* __restrict__ out) {
  int n = blockIdx.x * blockDim.x + threadIdx.x;
  if (n >= NSEQ) return;
  const float4* av = (const float4*)(a + (long)n * NHID);
  const float4* wv = (const float4*)w;
  float s = 0.f;
  #pragma unroll
  for (int q = 0; q < 16; ++q) {
    float4 x = av[q], y = wv[q];
    s += x.x * y.x + x.y * y.y + x.z * y.z + x.w * y.w;
  }
  out[n] = fmaxf(s + b[0], 0.f);
}

// ------------------------------------------------------------------- driver
static inline int divup_(long a, long b) { return (int)((a + b - 1) / b); }

extern "C" void kernel_launch(void* const* d_in, const int* in_sizes, int n_in,
                              void* d_out, int out_size, void* d_ws, size_t ws_size,
                              hipStream_t stream) {
  const float* x     = (const float*)d_in[0];
  const int*   ei    = (const int*)  d_in[1];
  const float* ew    = (const float*)d_in[2];
  const float* c1w   = (const float*)d_in[3];
  const float* c1b   = (const float*)d_in[4];
  const float* c2w   = (const float*)d_in[5];
  const float* c2b   = (const float*)d_in[6];
  const float* bn1g  = (const float*)d_in[7];
  const float* bn1b  = (const float*)d_in[8];
  const float* bn2g  = (const float*)d_in[9];
  const float* bn2b  = (const float*)d_in[10];
  const float* l1wih = (const float*)d_in[11];
  const float* l1whh = (const float*)d_in[12];
  const float* l1bih = (const float*)d_in[13];
  const float* l1bhh = (const float*)d_in[14];
  const float* l2wih = (const float*)d_in[15];
  const float* l2whh = (const float*)d_in[16];
  const float* l2bih = (const float*)d_in[17];
  const float* l2bhh = (const float*)d_in[18];
  const float* fc1w  = (const float*)d_in[19];
  const float* fc1b  = (const float*)d_in[20];
  const float* fc2w  = (const float*)d_in[21];
  const float* fc2b  = (const float*)d_in[22];
  float* out = (float*)d_out;

  // ---- workspace layout (floats); big regions are reused phase-to-phase
  const long NTH = (long)NTOT * NHID;          // 25.6M
  float* ws   = (float*)d_ws;
  float* xl   = ws;                            // R0: GCN linear out
  float* agg  = ws + NTH;                      // R1: GCN aggregation
  float* h1   = ws + 2 * NTH;                  // persists
  float* h2   = ws + 3 * NTH;                  // persists
  float* misc = ws + 4 * NTH;
  float*  deg   = misc;                        // NTOT (becomes dinv in place)
  double* stats = (double*)(misc + NTOT);      // 128 doubles (8B aligned)
  float*  wcat1 = (float*)(stats + 128);       // 192*256
  float*  bcat1 = wcat1 + 192 * 256;           // 256
  float*  wcat2 = bcat1 + 256;                 // 128*256
  float*  bcat2 = wcat2 + 128 * 256;           // 256
  float*  fc1wT = bcat2 + 256;                 // 208*64
  // phase reuse: R0 -> gates + Astep ; R1 -> LSTM states + Z + fc1out
  float* gates  = xl;                          // NSEQ*256
  float* astep  = xl + (long)NSEQ * 256;       // NSEQ*192
  float* h1s    = agg;                         // NSEQ*64
  float* c1s    = agg + 1L * NSEQ * NHID;
  float* h2s    = agg + 2L * NSEQ * NHID;
  float* c2s    = agg + 3L * NSEQ * NHID;
  float* Z      = agg + 4L * NSEQ * NHID;      // NSEQ*208
  float* fc1out = Z + (long)NSEQ * 208;        // NSEQ*64

  auto gemm = [&](const float* A, const float* W, const float* bias, float* C,
                  int M, int K, int N, int relu) {
    dim3 grid(divup_(M / 32, 8), N / 64);
    k_gemm<<<grid, 256, 0, stream>>>(A, W, bias, C, M, K, N, relu);
  };

  // ---- weight packing (independent of data)
  k_pack_lstm_w<<<divup_(192 * 256, 256), 256, 0, stream>>>(l1wih, l1whh, l1bih, l1bhh, wcat1, bcat1, 128);
  k_pack_lstm_w<<<divup_(128 * 256, 256), 256, 0, stream>>>(l2wih, l2whh, l2bih, l2bhh, wcat2, bcat2, 64);
  k_pack_fc1w<<<divup_(208 * 64, 256), 256, 0, stream>>>(fc1w, fc1wT);

  // ---- degree + symmetric normalization (self loop weight 1)
  k_fill<<<divup_(NTOT, 256), 256, 0, stream>>>(deg, 1.0f, NTOT);
  k_deg<<<divup_(NEDGE, 256), 256, 0, stream>>>(ei + NEDGE, ew, deg);
  k_dinv<<<divup_(NTOT, 256), 256, 0, stream>>>(deg);

  const long ESC = (long)NEDGE * 16;
  // ---- GCN layer 1
  gemm(x, c1w, nullptr, xl, NTOT, NFEAT, NHID, 0);
  k_selfloop<<<divup_(NTH, 256), 256, 0, stream>>>(xl, deg, agg);
  k_edge_scatter<<<divup_(ESC, 256), 256, 0, stream>>>(ei, ei + NEDGE, ew, deg, xl, agg);
  k_fill<<<1, 256, 0, stream>>>((float*)stats, 0.0f, 256);
  k_bias_relu_stats<<<1024, 256, 0, stream>>>(agg, c1b, h1, stats);
  k_bn_apply<<<divup_(NTH, 256), 256, 0, stream>>>(h1, stats, bn1g, bn1b);

  // ---- GCN layer 2
  gemm(h1, c2w, nullptr, xl, NTOT, NHID, NHID, 0);
  k_selfloop<<<divup_(NTH, 256), 256, 0, stream>>>(xl, deg, agg);
  k_edge_scatter<<<divup_(ESC, 256), 256, 0, stream>>>(ei, ei + NEDGE, ew, deg, xl, agg);
  k_fill<<<1, 256, 0, stream>>>((float*)stats, 0.0f, 256);
  k_bias_relu_stats<<<1024, 256, 0, stream>>>(agg, c2b, h2, stats);
  k_bn_apply<<<divup_(NTH, 256), 256, 0, stream>>>(h2, stats, bn2g, bn2b);
  // (agg/xl regions free now -> reused as LSTM state/gates buffers)

  // ---- stacked LSTM over WIN steps (layer-2 consumes layer-1 output per step)
  k_fill<<<divup_(4L * NSEQ * NHID, 256), 256, 0, stream>>>(h1s, 0.0f, 4L * NSEQ * NHID);
  for (int t = 0; t < WINLEN; ++t) {
    k_gather_lstm1<<<NSEQ, 192, 0, stream>>>(h1, h2, h1s, astep, t);
    gemm(astep, wcat1, bcat1, gates, NSEQ, 192, 256, 0);
    k_lstm_cell<<<divup_((long)NSEQ * NHID, 256), 256, 0, stream>>>(gates, c1s, h1s);
    k_gather_lstm2<<<NSEQ, 128, 0, stream>>>(h1s, h2s, astep);
    gemm(astep, wcat2, bcat2, gates, NSEQ, 128, 256, 0);
    k_lstm_cell<<<divup_((long)NSEQ * NHID, 256), 256, 0, stream>>>(gates, c2s, h2s);
  }

  // ---- head: z = [hn1|hn2|skip] -> relu(fc1) -> relu(fc2)
  k_gather_z<<<NSEQ, 256, 0, stream>>>(h1s, h2s, x, Z);
  gemm(Z, fc1wT, fc1b, fc1out, NSEQ, 208, NHID, 1);
  k_fc2<<<divup_(NSEQ, 256), 256, 0, stream>>>(fc1out, fc2w, fc2b, out);
}